// LSTM_29755533427140
// MI455X (gfx1250) — compile-verified
//
#include <hip/hip_runtime.h>
#include <hip/hip_bf16.h>

typedef float v2f __attribute__((ext_vector_type(2)));
typedef float v8f __attribute__((ext_vector_type(8)));

#define T_LEN    168
#define BM       128      // batch rows per block
#define H_STRIDE 172      // padded LDS row stride (floats): even (b64-aligned), low-conflict
#define DOUT     48
#define THREADS  128

__device__ __forceinline__ float fast_sigmoid(float x) {
    return __builtin_amdgcn_rcpf(1.0f + __expf(-x));
}
__device__ __forceinline__ float fast_tanh(float x) {
    return 2.0f * __builtin_amdgcn_rcpf(1.0f + __expf(-2.0f * x)) - 1.0f;
}

__global__ __launch_bounds__(THREADS)
void lstm_scan_wmma_kernel(const float* __restrict__ x,      // [B, T, 1]
                           const float* __restrict__ lk,     // [2, 4] gate order i,j,f,o
                           const float* __restrict__ lb,     // [4]
                           const float* __restrict__ w,      // [T, 48]
                           const float* __restrict__ ob,     // [48]
                           float* __restrict__ out)          // [B, 48]
{
    __shared__ float h_lds[BM * H_STRIDE];     // holds x tile, overwritten by h in place
    __shared__ float w_lds[T_LEN * DOUT];      // readout weights, stride 48

    const int tid = threadIdx.x;
    const long long blockRow0 = (long long)blockIdx.x * BM;

    // ---- stage x tile: 128 rows x 168 f32, contiguous in global -> coalesced float4 ----
    const float* xblk = x + blockRow0 * T_LEN;
    for (int i = tid; i < (BM * T_LEN) / 4; i += THREADS) {
        float4 v  = ((const float4*)xblk)[i];
        int g0    = i * 4;
        int row   = g0 / T_LEN;
        int col   = g0 - row * T_LEN;           // T_LEN % 4 == 0 -> never crosses a row
        *(float4*)&h_lds[row * H_STRIDE + col] = v;
    }
    // ---- stage W (168*48 f32 contiguous) ----
    for (int i = tid; i < (T_LEN * DOUT) / 4; i += THREADS) {
        ((float4*)w_lds)[i] = ((const float4*)w)[i];
    }

    // uniform gate weights -> scalar registers
    const float kxi = lk[0], kxj = lk[1], kxf = lk[2], kxo = lk[3];
    const float khi = lk[4], khj = lk[5], khf = lk[6], kho = lk[7];
    const float bi  = lb[0], bj  = lb[1], bf  = lb[2], bo  = lb[3];

    __syncthreads();

    // ---- phase 1: serial LSTM scan, one thread per batch row; write h over x ----
    {
        float c = 0.0f, h = 0.0f;
        float* row = &h_lds[tid * H_STRIDE];
        #pragma unroll 4
        for (int t = 0; t < T_LEN; ++t) {
            float xv = row[t];
            float zi = fmaf(xv, kxi, fmaf(h, khi, bi));
            float zj = fmaf(xv, kxj, fmaf(h, khj, bj));
            float zf = fmaf(xv, kxf, fmaf(h, khf, bf));
            float zo = fmaf(xv, kxo, fmaf(h, kho, bo));
            c = fast_sigmoid(zf + 1.0f) * c + fast_sigmoid(zi) * fast_tanh(zj);
            h = fast_sigmoid(zo) * fast_tanh(c);
            row[t] = h;
        }
    }

    __syncthreads();

    // ---- phase 2: out[128 x 48] = h[128 x 168] @ W[168 x 48] + ob, via f32 WMMA ----
    const int wave  = tid >> 5;         // 4 waves
    const int lane  = tid & 31;
    const int laneM = lane & 15;
    const int hi    = lane >> 4;        // lanes 16-31 carry K+2 half of the fragment

    // 8 M-tiles x 3 N-tiles = 24 tiles, 6 per wave (wave-uniform loop -> EXEC all ones)
    for (int tile = wave; tile < 24; tile += 4) {
        const int mt = tile / 3, nt = tile - mt * 3;
        const int m0 = mt * 16, n0 = nt * 16;

        const float biasv = ob[n0 + laneM];   // C/D: N = lane & 15 for all 8 VGPRs
        v8f acc;
        #pragma unroll
        for (int v = 0; v < 8; ++v) acc[v] = biasv;

        const float* arow = &h_lds[(m0 + laneM) * H_STRIDE];

        #pragma unroll 2
        for (int k0 = 0; k0 < T_LEN; k0 += 4) {
            // A 16x4: lanes 0-15 -> (M=laneM, K=k0..k0+1), lanes 16-31 -> (M=laneM, K=k0+2..k0+3)
            v2f a = *(const v2f*)&arow[k0 + 2 * hi];          // 8B-aligned ds_load_b64
            // B 4x16 (mirrors A): v0 = rows K=k0 / k0+2, v1 = rows K=k0+1 / k0+3, N = laneM
            v2f b;
            b.x = w_lds[(k0 + 2 * hi)     * DOUT + laneM];
            b.y = w_lds[(k0 + 2 * hi + 1) * DOUT + laneM];
            acc = __builtin_amdgcn_wmma_f32_16x16x4_f32(
                      /*neg_a=*/false, a, /*neg_b=*/false, b,
                      /*c_mod=*/(short)0, acc, /*reuse_a=*/false, /*reuse_b=*/false);
        }

        // D layout: VGPR v, lanes 0-15 -> M = v, lanes 16-31 -> M = v + 8; N = laneM
        float* obase = out + (blockRow0 + m0 + 8 * hi) * DOUT + n0 + laneM;
        #pragma unroll
        for (int v = 0; v < 8; ++v) {
            obase[(long long)v * DOUT] = acc[v];
        }
    }
}

extern "C" void kernel_launch(void* const* d_in, const int* in_sizes, int n_in,
                              void* d_out, int out_size, void* d_ws, size_t ws_size,
                              hipStream_t stream) {
    (void)n_in; (void)out_size; (void)d_ws; (void)ws_size;
    const float* x  = (const float*)d_in[0];   // [B, T, 1]
    const float* lk = (const float*)d_in[1];   // [2, 4]
    const float* lb = (const float*)d_in[2];   // [4]
    const float* w  = (const float*)d_in[3];   // [T, 48]
    const float* ob = (const float*)d_in[4];   // [48]
    float* out = (float*)d_out;                // [B, 48]

    const int rows    = in_sizes[0] / T_LEN;   // B = 131072
    const int nBlocks = rows / BM;             // 1024

    lstm_scan_wmma_kernel<<<nBlocks, THREADS, 0, stream>>>(x, lk, lb, w, ob, out);
}